// encoder_layer_2d_1to1_14946486190134
// MI455X (gfx1250) — compile-verified
//
#include <hip/hip_runtime.h>

// LIF temporal scan recast as triangular matmul on the CDNA5 WMMA pipe.
//   V^T[p,t] = sum_k Xs^T[p,k] * U[k,t],  U[k][t] = decay^(t-k) (t>=k)
//   Xs[p,k]  = gain[p]*weight[p]*(1-decay) * x[p,k]
// spikes = V > 1;  final_v = V[:,49];  final_r[p] = sum_t decay^(49-t)*spike[p,t]
// bias input is unused by the reference.
//
// decay powers are factored as decay^(4e) * decay^(n-2*khalf) * decay^(-j):
// the first factor is a compile-time constant table, the lane-dependent part
// is ONE raw v_exp_f32 (__builtin_amdgcn_exp2f) -- no libm range-fixup code.

typedef __attribute__((ext_vector_type(2))) float v2f;
typedef __attribute__((ext_vector_type(8))) float v8f;

#define TT    50
#define CHW   (3 * 128 * 128)
#define NPIX  (32 * CHW)
// decay = exp(-0.25);  -log2(decay) = 0.25/ln(2)
#define NL2D     0.36067376022224085f
#define OMD      0.221199216928595f    // 1 - exp(-0.25)
#define INV_DEC  1.2840254166877414f   // exp(+0.25) = decay^-1

// Ce[e+3] = decay^(4e) = exp(-e), e = -3..12
__constant__ const float CE_TAB[16] = {
    20.085536923187668f, 7.38905609893065f, 2.718281828459045f, 1.0f,
    0.36787944117144233f, 0.1353352832366127f, 0.049787068367863944f,
    0.018315638888734179f, 0.006737946999085467f, 0.0024787521766663585f,
    0.0009118819655545162f, 0.00033546262790251185f, 0.00012340980408667956f,
    4.5399929762484854e-05f, 1.670170079024566e-05f, 6.14421235332821e-06f };

// Kt[tn] = decay^(49-16*tn) = exp(-0.25*(49-16*tn))
__constant__ const float KT_TAB[4] = {
    4.785117392129009e-06f, 2.6125855730166754e-04f,
    1.4264233908999256e-02f, 0.7788007830714049f };

__global__ __launch_bounds__(128) void lif_scan_wmma_kernel(
    const float* __restrict__ x,      // [NPIX, TT]
    const float* __restrict__ wgt,    // [CHW]
    const float* __restrict__ gin,    // [CHW]
    float* __restrict__ spikes,       // [NPIX, TT]
    float* __restrict__ fv,           // [NPIX]
    float* __restrict__ fr)           // [NPIX]
{
    const int lane  = threadIdx.x & 31;
    const int wave  = blockIdx.x * (blockDim.x >> 5) + (threadIdx.x >> 5);
    const int P0    = wave * 16;          // 16 pixels per wave
    const int m     = lane & 15;          // A/D row id (pixel within tile)
    const int khalf = lane >> 4;          // which K half this lane holds
    const int n     = lane & 15;          // B/D column id (time within tile)

    // Per-pixel input scale (params broadcast over batch)
    const int   pm    = P0 + m;
    const int   q     = pm % CHW;
    const float scale = gin[q] * wgt[q] * OMD;

    // ---- Load A tiles (streaming, non-temporal): 13 chunks of K=4.
    // Lane layout for f32 16x16x4 A: lanes 0-15 hold {k0,k1}, 16-31 hold {k2,k3}.
    const float* xp = x + (size_t)pm * TT;
    v2f a[13];
#pragma unroll
    for (int kc = 0; kc < 12; ++kc)
        a[kc] = __builtin_nontemporal_load((const v2f*)(xp + kc * 4 + khalf * 2));
    // chunk 12: k = 48,49 valid (khalf 0); k = 50,51 padded zero (khalf 1)
    if (khalf == 0) a[12] = __builtin_nontemporal_load((const v2f*)(xp + 48));
    else            { v2f z = {0.0f, 0.0f}; a[12] = z; }
#pragma unroll
    for (int kc = 0; kc < 13; ++kc) a[kc] *= scale;

    // ---- B (decay) tiles: 16 distinct tiles indexed by e = 4*tn - kc.
    // Element (j in {0,1}): decay^(d), d = 4e + n - 2*khalf - j, zero if d<0.
    // Lane-dependent factor: one raw v_exp_f32.
    const float g0 = __builtin_amdgcn_exp2f(-NL2D * (float)(n - 2 * khalf)); // decay^(n-2khalf)
    const float g1 = g0 * INV_DEC;                                           // decay^(n-2khalf-1)
    v2f bt[16];
#pragma unroll
    for (int e = -3; e <= 12; ++e) {
        const float Ce = CE_TAB[e + 3];
        v2f b;
        if (e >= 1) {               // d0 >= 2 for all lanes: no masking needed
            b.x = Ce * g0;
            b.y = Ce * g1;
        } else {
            const int d0 = 4 * e + n - 2 * khalf;
            b.x = (d0 >= 0) ? Ce * g0 : 0.0f;
            b.y = (d0 >= 1) ? Ce * g1 : 0.0f;
        }
        bt[e + 3] = b;
    }

    // ---- Triangular accumulation: skip all-zero upper chunks (kc > 4*tn+3)
    v8f acc[4];
#pragma unroll
    for (int tn = 0; tn < 4; ++tn) {
        v8f c = {};
        const int kmax = (4 * tn + 3 > 12) ? 12 : 4 * tn + 3;
#pragma unroll
        for (int kc = 0; kc <= kmax; ++kc) {
            c = __builtin_amdgcn_wmma_f32_16x16x4_f32(
                    false, a[kc], false, bt[4 * tn + 3 - kc],
                    (short)0, c, false, false);
        }
        acc[tn] = c;
    }

    // ---- Threshold, store spikes (non-temporal), reduce final_v / final_r
    const float hinv = __builtin_amdgcn_exp2f(NL2D * (float)n);  // decay^(-n)
    float frr[8];
#pragma unroll
    for (int r = 0; r < 8; ++r) frr[r] = 0.0f;

#pragma unroll
    for (int tn = 0; tn < 4; ++tn) {
        float w49 = KT_TAB[tn] * hinv;              // decay^(49-t), t = 16*tn+n
        if (tn == 3 && n > 1) w49 = 0.0f;           // t > 49
#pragma unroll
        for (int r = 0; r < 8; ++r) {
            const int   pix = P0 + r + 8 * khalf;
            const float v   = acc[tn][r];
            const float s   = (v > 1.0f) ? 1.0f : 0.0f;
            if (tn < 3 || n < 2)                    // t < 50
                __builtin_nontemporal_store(s, spikes + (size_t)pix * TT + tn * 16 + n);
            frr[r] = fmaf(s, w49, frr[r]);
            if (tn == 3 && n == 1) fv[pix] = v;     // t == 49
        }
    }

    // 16-lane tree reduction within each half (xor masks stay inside the half)
#pragma unroll
    for (int r = 0; r < 8; ++r) {
        float v = frr[r];
        v += __shfl_xor(v, 1, 32);
        v += __shfl_xor(v, 2, 32);
        v += __shfl_xor(v, 4, 32);
        v += __shfl_xor(v, 8, 32);
        if (n == 0) fr[P0 + r + 8 * khalf] = v;
    }
}

extern "C" void kernel_launch(void* const* d_in, const int* in_sizes, int n_in,
                              void* d_out, int out_size, void* d_ws, size_t ws_size,
                              hipStream_t stream) {
    const float* x   = (const float*)d_in[0];  // input_spikes [B,C,H,W,T]
    const float* wgt = (const float*)d_in[1];  // weight [C,H,W]
    const float* gin = (const float*)d_in[2];  // gain   [C,H,W]
    // d_in[3] = bias: unused by the reference computation.

    float* out    = (float*)d_out;
    float* spikes = out;                          // NPIX*TT
    float* fv     = out + (size_t)NPIX * TT;      // NPIX
    float* fr     = fv + NPIX;                    // NPIX

    // 16 pixels per wave, 4 waves per block -> 64 pixels per block
    const int waves  = NPIX / 16;                 // 98,304
    const int blocks = waves / 4;                 // 24,576
    lif_scan_wmma_kernel<<<blocks, 128, 0, stream>>>(x, wgt, gin, spikes, fv, fr);
}